// DenseAssociativeEmbeddingLoss_78108275245640
// MI455X (gfx1250) — compile-verified
//
#include <hip/hip_runtime.h>
#include <hip/hip_bf16.h>

typedef __attribute__((ext_vector_type(2))) float v2f;
typedef __attribute__((ext_vector_type(8))) float v8f;

#define BB 16
#define CC 8
#define HW (512 * 512)
#define NN 32
#define PP 64
#define MARGIN_F 2.0f

__device__ __forceinline__ float wave_reduce_add(float v) {
    // deterministic fixed xor-tree over wave32
    v += __shfl_xor(v, 16, 32);
    v += __shfl_xor(v, 8, 32);
    v += __shfl_xor(v, 4, 32);
    v += __shfl_xor(v, 2, 32);
    v += __shfl_xor(v, 1, 32);
    return v;
}

// K1: one wave32 per (b, n) instance.
// Computes S_c = sum_p g[p][c] and S2_c = sum_p g[p][c]^2 for c=0..7 using
// chained V_WMMA_F32_16X16X4_F32 with A = ones(16x4):
//   B columns 0..7  = g values  (4 K-points per WMMA, split across lane halves)
//   B columns 8..15 = g^2 values
// Every D row then holds the 16 column sums, so acc[0] in lane l is the sum
// for column (l & 15).
__global__ __launch_bounds__(32)
void ae_inst_kernel(const float* __restrict__ pred, const int* __restrict__ inds,
                    float* __restrict__ s_ws, float* __restrict__ pull_ws) {
    const int inst = blockIdx.x;        // 0..B*N-1
    const int b    = inst >> 5;
    const int n    = inst & (NN - 1);
    const int lane = threadIdx.x;       // 0..31
    const int col  = lane & 15;         // WMMA column this lane feeds (N dim)
    const int half = lane >> 4;         // lane half selects K pair {0,1} vs {2,3}
    const int chan = col & 7;
    const bool isSq = (col & 8) != 0;

    const float* __restrict__ predb = pred + ((size_t)b * CC + (size_t)chan) * (size_t)HW;
    const int*   __restrict__ indb  = inds + ((size_t)b * NN + n) * PP;

#if __has_builtin(__builtin_amdgcn_wmma_f32_16x16x4_f32)
    v8f acc = {};
    v2f ones;
    ones.x = 1.0f;
    ones.y = 1.0f;
#pragma unroll 4
    for (int it = 0; it < PP / 4; ++it) {
        const int p0 = it * 4 + half * 2;
        const int iA = indb[p0];
        const int iB = indb[p0 + 1];
        const float vA = predb[iA];
        const float vB = predb[iB];
        v2f bvec;                       // VGPR0 = B[K=2h][col], VGPR1 = B[K=2h+1][col]
        bvec.x = isSq ? vA * vA : vA;
        bvec.y = isSq ? vB * vB : vB;
        acc = __builtin_amdgcn_wmma_f32_16x16x4_f32(
            /*neg_a=*/false, ones, /*neg_b=*/false, bvec,
            /*c_mod=*/(short)0, acc, /*reuse_a=*/false, /*reuse_b=*/false);
    }
    float colsum = acc[0];              // all D rows identical -> per-column sum
#else
    // Scalar fallback (non-WMMA toolchains): same data placement, explicit K sum.
    float colsum = 0.0f;
    for (int it = 0; it < PP / 4; ++it) {
        const int p0 = it * 4 + half * 2;
        const int iA = indb[p0];
        const int iB = indb[p0 + 1];
        const float vA = predb[iA];
        const float vB = predb[iB];
        colsum += (isSq ? vA * vA : vA) + (isSq ? vB * vB : vB);
    }
    colsum += __shfl_xor(colsum, 16, 32);  // combine the two K-halves per column
#endif

    // Lane c (c<8) holds S_c; lane c+8 holds S2_c. Pull per instance:
    //   (1/P) * sum_c (S2_c - S_c^2 / P);    s = (1/P) * sum_c S_c
    const float sg   = colsum;
    const float sg2  = __shfl(colsum, col + 8, 32);   // valid where col < 8
    float pullv = (lane < 8) ? (sg2 - sg * sg * (1.0f / (float)PP)) : 0.0f;
    float sv    = (lane < 8) ? sg : 0.0f;
    pullv = wave_reduce_add(pullv);
    sv    = wave_reduce_add(sv);
    if (lane == 0) {
        pull_ws[inst] = pullv * (1.0f / (float)PP);
        s_ws[inst]    = sv * (1.0f / (float)PP);
    }
}

// K2: one wave32 per image b; full 32x32 pairwise margin via lane broadcasts.
__global__ __launch_bounds__(32)
void ae_push_kernel(const float* __restrict__ s_ws, float* __restrict__ push_ws) {
    const int b = blockIdx.x;
    const int i = threadIdx.x;
    const float si = s_ws[b * NN + i];
    float conf = 0.0f;
#pragma unroll
    for (int j = 0; j < NN; ++j) {
        const float sj = __shfl(si, j, 32);
        const float r  = fmaxf(MARGIN_F - fabsf(si - sj), 0.0f);
        conf += (j == i) ? 0.0f : r;
    }
    conf = wave_reduce_add(conf);
    if (i == 0) push_ws[b] = conf;
}

// K3: single wave, deterministic final reduction into d_out[0..1].
__global__ __launch_bounds__(32)
void ae_final_kernel(const float* __restrict__ pull_ws, const float* __restrict__ push_ws,
                     float* __restrict__ out) {
    const int lane = threadIdx.x;
    float p = 0.0f;
#pragma unroll
    for (int k = 0; k < (BB * NN) / 32; ++k) p += pull_ws[k * 32 + lane];
    p = wave_reduce_add(p);
    float q = (lane < BB) ? push_ws[lane] : 0.0f;
    q = wave_reduce_add(q);
    if (lane == 0) {
        out[0] = 0.25f * p;
        out[1] = 0.25f * q / (float)(NN * (NN - 1));
    }
}

extern "C" void kernel_launch(void* const* d_in, const int* in_sizes, int n_in,
                              void* d_out, int out_size, void* d_ws, size_t ws_size,
                              hipStream_t stream) {
    const float* pred = (const float*)d_in[0];   // [B, C, H, W] f32
    const int*   inds = (const int*)d_in[1];     // [B, N, P] int
    float* out = (float*)d_out;                  // [2] f32

    float* s_ws    = (float*)d_ws;               // B*N floats
    float* pull_ws = s_ws + BB * NN;             // B*N floats
    float* push_ws = pull_ws + BB * NN;          // B  floats

    ae_inst_kernel<<<BB * NN, 32, 0, stream>>>(pred, inds, s_ws, pull_ws);
    ae_push_kernel<<<BB, 32, 0, stream>>>(s_ws, push_ws);
    ae_final_kernel<<<1, 32, 0, stream>>>(pull_ws, push_ws, out);
}